// PV_RCNN_42417097016364
// MI455X (gfx1250) — compile-verified
//
#include <hip/hip_runtime.h>
#include <math.h>

typedef __attribute__((ext_vector_type(2))) float v2f;
typedef __attribute__((ext_vector_type(8))) float v8f;

#define Bc    4
#define NANCH 211200
#define Cc    256
#define Hc    200
#define Wc    176
#define Kc    4096
#define HWc   (Hc * Wc)

__device__ __forceinline__ int clampi(int v, int lo, int hi) {
    return v < lo ? lo : (v > hi ? hi : v);
}

// ---------------------------------------------------------------------------
// Kernel 1: VoxelNet-style box decode. Pure elementwise, HBM-bound (~71 MB).
// One thread per anchor; a wave covers 896 contiguous bytes per array.
// ---------------------------------------------------------------------------
__global__ __launch_bounds__(256) void decode_kernel(
    const float* __restrict__ deltas, const float* __restrict__ anchors,
    float* __restrict__ boxes, int total) {
    int i = blockIdx.x * blockDim.x + threadIdx.x;
    if (i >= total) return;
    const float* d = deltas + (size_t)i * 7;
    const float* a = anchors + (size_t)i * 7;
    float* o = boxes + (size_t)i * 7;
    float aw = a[3], al = a[4], ah = a[5];
    float dn = sqrtf(aw * aw + al * al);
    o[0] = d[0] * dn + a[0];
    o[1] = d[1] * dn + a[1];
    o[2] = d[2] * ah + a[2];
    o[3] = expf(d[3]) * aw;
    o[4] = expf(d[4]) * al;
    o[5] = expf(d[5]) * ah;
    o[6] = d[6] + a[6];
}

// ---------------------------------------------------------------------------
// Kernel 2: BEV bilinear gather as chained V_WMMA_F32_16X16X4_F32.
// One wave computes a 16-channel x 16-keypoint tile:
//   step t: A_t[c][j] = texel j of keypoint t at channel c (16x4, f32)
//           B_t[j][k] = (k==t) ? bilinear weight j (validity-masked) : 0
//   D += A_t * B_t   (exact f32 math, identical to scalar FMA blend)
// A layout: lanes 0-15 supply K=0,1 (y0 row taps), lanes 16-31 K=2,3 (y1 row).
// D layout: VGPR v / lane L -> channel cbase+v+8*(L>=16), keypoint kbase+(L&15)
//           => stores coalesce across lanes (keypoint is the fast axis of out).
// ---------------------------------------------------------------------------
__global__ __launch_bounds__(256) void bev_wmma_kernel(
    const float* __restrict__ fm, const float* __restrict__ kp,
    float* __restrict__ out) {
    const int lane = threadIdx.x & 31;
    const int wave = threadIdx.x >> 5;
    const int wg   = blockIdx.x * (blockDim.x >> 5) + wave;

    const int CG = Cc / 16;            // 16 channel groups
    const int KG = Kc / 16;            // 256 keypoint groups
    int b   = wg / (KG * CG);
    int r   = wg - b * (KG * CG);
    int kgi = r / CG;
    int cgi = r - kgi * CG;
    int kbase = kgi * 16;
    int cbase = cgi * 16;

    const int  t15 = lane & 15;
    const bool hi  = lane >= 16;

    // Per-lane: pixel coords for keypoint kbase + t15 (reference math, step by
    // step: world->pixel, clamp, normalize by (dim-2), axis flip, denormalize).
    const float* kpp = kp + ((size_t)b * Kc + (kbase + t15)) * 3;
    float x = kpp[0], y = kpp[1];
    float xp = (x - 0.0f)  * 2.5f;                 // / (0.05 * 8)
    float yp = (y + 40.0f) * 2.5f;
    xp = fminf(fmaxf(xp, 0.0f), (float)(Wc - 1));
    yp = fminf(fmaxf(yp, 0.0f), (float)(Hc - 1));
    float xn = 2.0f * (xp / (float)(Wc - 2)) - 1.0f;
    float yn = 2.0f * (yp / (float)(Hc - 2)) - 1.0f;
    float ix = (yn + 1.0f) * 0.5f * (float)(Wc - 1);   // flip: width <- y
    float iy = (xn + 1.0f) * 0.5f * (float)(Hc - 1);   // flip: height <- x

    // This lane's channel plane (row M = lane&15 of the A matrix).
    const float* plane = fm + ((size_t)b * Cc + (cbase + t15)) * HWc;

    v8f acc = {};
#pragma unroll
    for (int t = 0; t < 16; ++t) {
        // Broadcast keypoint t's sample point to the whole wave (wave32).
        float ixt = __shfl(ix, t, 32);
        float iyt = __shfl(iy, t, 32);
        float x0f = floorf(ixt), y0f = floorf(iyt);
        float wx = ixt - x0f, wy = iyt - y0f;
        int x0 = (int)x0f, y0 = (int)y0f;
        int x1 = x0 + 1,   y1 = y0 + 1;
        // Zero-padding validity (reference gathers outside can occur).
        float vx0 = (x0 >= 0 && x0 < Wc) ? 1.0f : 0.0f;
        float vx1 = (x1 >= 0 && x1 < Wc) ? 1.0f : 0.0f;
        float vy0 = (y0 >= 0 && y0 < Hc) ? 1.0f : 0.0f;
        float vy1 = (y1 >= 0 && y1 < Hc) ? 1.0f : 0.0f;
        float w00 = (1.0f - wx) * (1.0f - wy) * vx0 * vy0;
        float w01 = wx * (1.0f - wy) * vx1 * vy0;
        float w10 = (1.0f - wx) * wy * vx0 * vy1;
        float w11 = wx * wy * vx1 * vy1;
        int x0c = clampi(x0, 0, Wc - 1), x1c = clampi(x1, 0, Wc - 1);
        int y0c = clampi(y0, 0, Hc - 1), y1c = clampi(y1, 0, Hc - 1);

        // A: lo lanes gather the y0 row taps (K=0,1), hi lanes the y1 row (K=2,3).
        int yrow = (hi ? y1c : y0c) * Wc;
        v2f av;
        av.x = plane[yrow + x0c];
        av.y = plane[yrow + x1c];

        // B: one-hot column t carries the 4 bilinear weights.
        bool sel = (t15 == t);
        v2f bv;
        bv.x = sel ? (hi ? w10 : w00) : 0.0f;
        bv.y = sel ? (hi ? w11 : w01) : 0.0f;

        acc = __builtin_amdgcn_wmma_f32_16x16x4_f32(
            /*neg_a=*/false, av, /*neg_b=*/false, bv,
            /*c_mod=*/(short)0, acc, /*reuse_a=*/false, /*reuse_b=*/false);
    }

    // Store the 16x16 tile: lanes 0-15 and 16-31 each write 8 channel rows,
    // keypoint index striped across lanes -> 64B coalesced per half-wave.
    int kk  = kbase + t15;
    int chb = cbase + (hi ? 8 : 0);
    float* ob = out + ((size_t)b * Cc) * Kc + kk;
#pragma unroll
    for (int v = 0; v < 8; ++v) {
        ob[(size_t)(chb + v) * Kc] = acc[v];
    }
}

extern "C" void kernel_launch(void* const* d_in, const int* in_sizes, int n_in,
                              void* d_out, int out_size, void* d_ws, size_t ws_size,
                              hipStream_t stream) {
    const float* deltas  = (const float*)d_in[0];  // (B, N_ANCHORS, 7)
    const float* anchors = (const float*)d_in[1];  // (B, N_ANCHORS, 7)
    const float* fm      = (const float*)d_in[2];  // (B, C, H, W)
    const float* kp      = (const float*)d_in[3];  // (B, K, 3)

    float* boxes = (float*)d_out;                               // (B, N, 7)
    float* bev   = (float*)d_out + (size_t)Bc * NANCH * 7;      // (B, C, K)

    int total = Bc * NANCH;
    decode_kernel<<<(total + 255) / 256, 256, 0, stream>>>(deltas, anchors,
                                                           boxes, total);

    // One wave per 16-channel x 16-keypoint tile: 4 * 256 * 16 = 16384 waves.
    int waves  = Bc * (Kc / 16) * (Cc / 16);
    int blocks = waves / 8;                        // 8 waves (256 thr) / block
    bev_wmma_kernel<<<blocks, 256, 0, stream>>>(fm, kp, bev);
}